// Torsion_5454608466123
// MI455X (gfx1250) — compile-verified
//
#include <hip/hip_runtime.h>
#include <math.h>

#define TPB 256
// Prefetch distance for the index stream, in torsions (~1 MB of int4 ahead).
#define PF_DIST (64 * 1024)

// Native clang vector type: __builtin_nontemporal_load requires a real vector,
// not HIP's HIP_vector_type class.
typedef int v4i __attribute__((ext_vector_type(4)));

__global__ __launch_bounds__(TPB)
void torsion_phi_kernel(const float* __restrict__ coords,
                        const int*   __restrict__ tors,
                        float*       __restrict__ out,
                        int n)
{
    int i = blockIdx.x * TPB + threadIdx.x;
    if (i >= n) return;

    // gfx1250: keep the single-use index stream flowing ahead of execution
    // (emits global_prefetch_b8; speculative, safe to clamp at the tail).
    int pf = i + PF_DIST;
    if (pf < n) {
        __builtin_prefetch(tors + 4 * (size_t)pf, 0, 0);
    }

    // One b128 load for the 4 indices; non-temporal so this 128MB stream
    // does not evict the L2-resident 24MB coords array.
    v4i t = __builtin_nontemporal_load((const v4i*)tors + i);

    const float* p0 = coords + 3 * (size_t)t.x;
    const float* p1 = coords + 3 * (size_t)t.y;
    const float* p2 = coords + 3 * (size_t)t.z;
    const float* p3 = coords + 3 * (size_t)t.w;

    // Random gathers: default (RT) temporal hint -> stay cached in L2.
    float ax = p0[0], ay = p0[1], az = p0[2];
    float bx = p1[0], by = p1[1], bz = p1[2];
    float cx = p2[0], cy = p2[1], cz = p2[2];
    float dx = p3[0], dy = p3[1], dz = p3[2];

    // Bond vectors
    float b1x = bx - ax, b1y = by - ay, b1z = bz - az;
    float b2x = cx - bx, b2y = cy - by, b2z = cz - bz;
    float b3x = dx - cx, b3y = dy - cy, b3z = dz - cz;

    // n1 = b1 x b2 ; n2 = b2 x b3
    float n1x = b1y * b2z - b1z * b2y;
    float n1y = b1z * b2x - b1x * b2z;
    float n1z = b1x * b2y - b1y * b2x;

    float n2x = b2y * b3z - b2z * b3y;
    float n2y = b2z * b3x - b2x * b3z;
    float n2z = b2x * b3y - b2y * b3x;

    float d12 = n1x * n2x + n1y * n2y + n1z * n2z;
    float q1  = n1x * n1x + n1y * n1y + n1z * n1z;
    float q2  = n2x * n2x + n2y * n2y + n2z * n2z;

    // cos(phi) = dot / (|n1||n2|); reference's fp32 clip bounds round to +/-1.
    float cosv = d12 * rsqrtf(q1 * q2);
    cosv = fminf(fmaxf(cosv, -1.0f), 1.0f);
    float phi = acosf(cosv);

    // sign = sign(dot(n1, b3)); sign > 0 ? phi : -phi  (0/NaN -> -phi)
    float s = n1x * b3x + n1y * b3y + n1z * b3z;
    float res = (s > 0.0f) ? phi : -phi;

    // Single-use output stream: non-temporal store.
    __builtin_nontemporal_store(res, out + i);
}

extern "C" void kernel_launch(void* const* d_in, const int* in_sizes, int n_in,
                              void* d_out, int out_size, void* d_ws, size_t ws_size,
                              hipStream_t stream) {
    const float* coords = (const float*)d_in[0];
    const int*   tors   = (const int*)d_in[1];   // harness delivers integer inputs as int32
    float*       out    = (float*)d_out;

    int n = out_size;                            // number of torsions (= in_sizes[1] / 4)
    int blocks = (n + TPB - 1) / TPB;
    torsion_phi_kernel<<<blocks, TPB, 0, stream>>>(coords, tors, out, n);
}